// AtomAttentionDecoder_60636348284984
// MI455X (gfx1250) — compile-verified
//
#include <hip/hip_runtime.h>
#include <math.h>

// ---------------------------------------------------------------------------
// Problem constants (from the reference)
// ---------------------------------------------------------------------------
#define T_TOK   512
#define A_EMB   768
#define Q_EMB   128
#define C_EMB   128
#define P_EMB   16
#define NBLK    3
#define NHEAD   4
#define NQ      32
#define NK      128
#define NTOK    1536          // N = 3*T
#define DH      32            // Q_EMB / NUM_HEAD
#define KCHUNK  128           // K-panel staged in LDS per GEMM iteration

typedef float v2f __attribute__((ext_vector_type(2)));
typedef float v8f __attribute__((ext_vector_type(8)));

// Epilogue codes for the generic WMMA GEMM
#define EPI_NONE     0
#define EPI_SIGMOID  1   // C = sigmoid(acc+bias)
#define EPI_SILU     2   // C = silu(acc+bias)
#define EPI_MUL      3   // C = (acc+bias) * extra
#define EPI_ADD      4   // C = (acc+bias) + extra
#define EPI_ADA      5   // C = sigmoid(extra)*extra2 + acc   (adaLN combine)
#define EPI_MULADD   6   // C = (acc+bias)*extra + extra2     (gate + residual)

__device__ __forceinline__ float sigmf(float x) { return 1.0f / (1.0f + expf(-x)); }

__device__ __forceinline__ float wred_sum(float v) {
  for (int o = 16; o > 0; o >>= 1) v += __shfl_xor(v, o, 32);
  return v;
}
__device__ __forceinline__ float wred_max(float v) {
  for (int o = 16; o > 0; o >>= 1) v = fmaxf(v, __shfl_xor(v, o, 32));
  return v;
}

// ---------------------------------------------------------------------------
// fp32 WMMA GEMM with LDS-staged B panel.
//   C[M,Ncols] = epi(A[M,K] @ diag(kscale) @ B[K,Ncols] + bias, ...)
// Workgroup = 8 waves: one 16-wide B column panel shared via LDS, each wave
// owns one 16-row M tile (128 rows per workgroup). B chunk (128x16 = 8 KB) is
// cooperatively loaded with float4 (ds_store_b128) and reused by all 8 waves;
// next chunk is prefetched (global_prefetch_b8, WGP scope) while computing.
// kscale (per-k row scale on B) folds the adaLN diag(s_ln_w) for free at
// staging time, keeping the inner WMMA loop branch-free:
//   (cn * diag(s)) @ W  ==  cn @ (diag(s) @ W)
// CDNA5 f32 WMMA layouts (ISA 7.12.2):
//   A 16x4 : lane&15 = M row; K pair {0,1}/{2,3} selected by lane>>4
//   B 4x16 : lane&15 = N col; same K pair split
//   C 16x16: VGPR r -> rows r / r+8, col = lane&15
// arowmod != 0 implements jnp.tile(a,(3,1)) via row index modulo.
// ---------------------------------------------------------------------------
__global__ __launch_bounds__(256) void gemm_epi(
    const float* __restrict__ A, int lda, int arowmod,
    const float* __restrict__ kscale,
    const float* __restrict__ B, int ldb,
    const float* __restrict__ bias,
    const float* __restrict__ extra, const float* __restrict__ extra2,
    float* __restrict__ C, int M, int K, int Ncols, int epi) {
  __shared__ float Bs[KCHUNK][16];

  const int nblocks = Ncols >> 4;
  const int nb = blockIdx.x % nblocks;
  const int mb = blockIdx.x / nblocks;
  const int wave = threadIdx.x >> 5;
  const int lane = threadIdx.x & 31;
  const int row0 = mb * 128 + wave * 16;
  const int col0 = nb * 16;

  const int m  = lane & 15;        // A row / B,C col within tile
  const int k2 = (lane >> 4) << 1; // K sub-offset {0,2}

  int ar = row0 + m;
  if (arowmod) ar %= arowmod;
  const float* Arow = A + (size_t)ar * lda;

  // cooperative-load mapping: thread t stages 8 consecutive B floats of row
  // (t>>1); the per-k adaLN scale is applied here, once per chunk.
  const int t = threadIdx.x;
  const int lrow = t >> 1;
  const int lcol = (t & 1) << 3;

  v8f acc = {};
  for (int kc = 0; kc < K; kc += KCHUNK) {
    {
      const float sc = kscale ? kscale[kc + lrow] : 1.0f;
      const float* src = B + (size_t)(kc + lrow) * ldb + col0 + lcol;
      float4 f0 = *(const float4*)(src);
      float4 f1 = *(const float4*)(src + 4);
      f0.x *= sc; f0.y *= sc; f0.z *= sc; f0.w *= sc;
      f1.x *= sc; f1.y *= sc; f1.z *= sc; f1.w *= sc;
      *(float4*)(&Bs[lrow][lcol]) = f0;
      *(float4*)(&Bs[lrow][lcol + 4]) = f1;
    }
    __syncthreads();

    // prefetch next chunk (near caches) while this one is consumed
    if (kc + KCHUNK < K)
      __builtin_prefetch(B + (size_t)(kc + KCHUNK + lrow) * ldb + col0 + lcol, 0, 3);

#pragma unroll 8
    for (int k0 = 0; k0 < KCHUNK; k0 += 4) {
      const int kk = kc + k0 + k2;
      v2f av, bv;
      av.x = Arow[kk];
      av.y = Arow[kk + 1];
      bv.x = Bs[k0 + k2][m];
      bv.y = Bs[k0 + k2 + 1][m];
      acc = __builtin_amdgcn_wmma_f32_16x16x4_f32(false, av, false, bv,
                                                  (short)0, acc, false, false);
    }
    __syncthreads();
  }

  const int col = col0 + m;
  const float bb = bias ? bias[col] : 0.0f;
  const int rbase = row0 + ((lane >> 4) << 3);
#pragma unroll
  for (int r = 0; r < 8; ++r) {
    const size_t idx = (size_t)(rbase + r) * Ncols + col;
    float v = acc[r] + bb;
    if (epi == EPI_SIGMOID)      v = sigmf(v);
    else if (epi == EPI_SILU)    v = v * sigmf(v);
    else if (epi == EPI_MUL)     v *= extra[idx];
    else if (epi == EPI_ADD)     v += extra[idx];
    else if (epi == EPI_ADA)     v = sigmf(extra[idx]) * extra2[idx] + v;
    else if (epi == EPI_MULADD)  v = v * extra[idx] + extra2[idx];
    C[idx] = v;
  }
}

// ---------------------------------------------------------------------------
// Row-wise non-affine LayerNorm over 128 columns, one wave per row
// ---------------------------------------------------------------------------
__global__ __launch_bounds__(256) void ln0_rows(const float* __restrict__ x,
                                                float* __restrict__ y) {
  const int wave = threadIdx.x >> 5, lane = threadIdx.x & 31;
  const int row = blockIdx.x * 8 + wave;
  if (row >= NTOK) return;
  const float* xr = x + (size_t)row * 128;
  float v[4], s = 0.f, s2 = 0.f;
#pragma unroll
  for (int c = 0; c < 4; ++c) {
    v[c] = xr[lane + 32 * c];
    s += v[c]; s2 += v[c] * v[c];
  }
  s = wred_sum(s); s2 = wred_sum(s2);
  const float mu = s * (1.0f / 128.0f);
  const float var = s2 * (1.0f / 128.0f) - mu * mu;
  const float rs = rsqrtf(var + 1e-5f);
  float* yr = y + (size_t)row * 128;
#pragma unroll
  for (int c = 0; c < 4; ++c) yr[lane + 32 * c] = (v[c] - mu) * rs;
}

// ---------------------------------------------------------------------------
// Windowed pair bias: bias[q][j][h] = LN(p[q, kstart(q)+j, :16]) . wb[:,h]
// Only the N*128 (q,key) pairs inside the attention window are touched
// (12.6 MB of p per block instead of 151 MB).
// ---------------------------------------------------------------------------
__global__ __launch_bounds__(256) void pair_bias_kernel(
    const float* __restrict__ p, const float* __restrict__ plw,
    const float* __restrict__ plb, const float* __restrict__ wb,
    float* __restrict__ bias) {
  const int idx = blockIdx.x * blockDim.x + threadIdx.x; // < NTOK*128
  const int q = idx >> 7, j = idx & 127;
  const int kstart = (q >> 5) * NQ - 48;
  const int k = kstart + j;
  float* out = bias + (size_t)idx * 4;
  if (k < 0 || k >= NTOK) {
    out[0] = out[1] = out[2] = out[3] = 0.0f;
    return;
  }
  const float* pr = p + ((size_t)q * NTOK + k) * P_EMB;
  float e[P_EMB], s = 0.f, s2 = 0.f;
#pragma unroll
  for (int tt = 0; tt < P_EMB; ++tt) { e[tt] = pr[tt]; s += e[tt]; s2 += e[tt] * e[tt]; }
  const float mu = s * (1.0f / P_EMB);
  const float var = s2 * (1.0f / P_EMB) - mu * mu;
  const float rs = rsqrtf(var + 1e-5f);
#pragma unroll
  for (int h = 0; h < 4; ++h) {
    float acc = 0.f;
#pragma unroll
    for (int tt = 0; tt < P_EMB; ++tt)
      acc += ((e[tt] - mu) * rs * plw[tt] + plb[tt]) * wb[tt * 4 + h];
    out[h] = acc;
  }
}

// ---------------------------------------------------------------------------
// Fused block-local attention: one workgroup per (query-block, head).
// QK^T (WMMA) -> logits+bias+mask in LDS -> softmax -> P*V (WMMA) -> gate.
// ---------------------------------------------------------------------------
__global__ __launch_bounds__(256) void attn_kernel(
    const float* __restrict__ qh, const float* __restrict__ kh,
    const float* __restrict__ vh, const float* __restrict__ gh,
    const float* __restrict__ bias, float* __restrict__ o) {
  __shared__ float lds[NQ][NK + 4];
  const int qb = blockIdx.x, h = blockIdx.y;
  const int q0 = qb * NQ;
  const int kstart = q0 - 48;
  const int wave = threadIdx.x >> 5, lane = threadIdx.x & 31;
  const int m = lane & 15, k2 = (lane >> 4) << 1;
  const float scale = 0.17677669529663687f; // 1/sqrt(32)

  // Phase 1: logits = Q K^T / sqrt(DH) + pair_bias  (16 tiles of 16x16)
  for (int t = wave; t < 16; t += 8) {
    const int tm = t >> 3, tn = t & 7;
    const float* qrow = qh + (size_t)(q0 + tm * 16 + m) * 128 + h * DH;
    v8f acc = {};
    const int key = kstart + tn * 16 + m;
    const bool kv = (key >= 0) && (key < NTOK);
    const float* krow = kh + (size_t)(kv ? key : 0) * 128 + h * DH;
#pragma unroll
    for (int k0 = 0; k0 < DH; k0 += 4) {
      v2f av, bv;
      av.x = qrow[k0 + k2];
      av.y = qrow[k0 + k2 + 1];
      bv.x = kv ? krow[k0 + k2] : 0.0f;
      bv.y = kv ? krow[k0 + k2 + 1] : 0.0f;
      acc = __builtin_amdgcn_wmma_f32_16x16x4_f32(false, av, false, bv,
                                                  (short)0, acc, false, false);
    }
    const int cl = tn * 16 + m;
#pragma unroll
    for (int r = 0; r < 8; ++r) {
      const int rl = tm * 16 + r + ((lane >> 4) << 3);
      float lg = -1e30f;
      if (kv)
        lg = acc[r] * scale + bias[((size_t)(q0 + rl) * 128 + cl) * 4 + h];
      lds[rl][cl] = lg;
    }
  }
  __syncthreads();

  // Phase 2: softmax over the 128 keys, 4 rows per wave
#pragma unroll
  for (int rr = 0; rr < 4; ++rr) {
    const int row = wave * 4 + rr;
    float v[4];
#pragma unroll
    for (int c = 0; c < 4; ++c) v[c] = lds[row][lane + 32 * c];
    float mx = fmaxf(fmaxf(v[0], v[1]), fmaxf(v[2], v[3]));
    mx = wred_max(mx);
    float s = 0.f;
#pragma unroll
    for (int c = 0; c < 4; ++c) { v[c] = expf(v[c] - mx); s += v[c]; }
    s = wred_sum(s);
    const float inv = 1.0f / s;
#pragma unroll
    for (int c = 0; c < 4; ++c) lds[row][lane + 32 * c] = v[c] * inv;
  }
  __syncthreads();

  // Phase 3: O = P V, then sigmoid-gate (4 tiles; waves 0-3)
  if (wave < 4) {
    const int tm = wave >> 1, tn = wave & 1;
    v8f acc = {};
    for (int k0 = 0; k0 < NK; k0 += 4) {
      v2f av, bv;
      av.x = lds[tm * 16 + m][k0 + k2];
      av.y = lds[tm * 16 + m][k0 + k2 + 1];
      const int key0 = kstart + k0 + k2;
      const int key1 = key0 + 1;
      bv.x = (key0 >= 0 && key0 < NTOK)
                 ? vh[(size_t)key0 * 128 + h * DH + tn * 16 + m] : 0.0f;
      bv.y = (key1 >= 0 && key1 < NTOK)
                 ? vh[(size_t)key1 * 128 + h * DH + tn * 16 + m] : 0.0f;
      acc = __builtin_amdgcn_wmma_f32_16x16x4_f32(false, av, false, bv,
                                                  (short)0, acc, false, false);
    }
    const int col = h * DH + tn * 16 + m;
#pragma unroll
    for (int r = 0; r < 8; ++r) {
      const int row = q0 + tm * 16 + r + ((lane >> 4) << 3);
      const float g = sigmf(gh[(size_t)row * 128 + col]);
      o[(size_t)row * 128 + col] = g * acc[r];
    }
  }
}

// ---------------------------------------------------------------------------
// Output head: LN(q, w, b) @ head_w[128,3] + head_b  -> out[N,3]
// ---------------------------------------------------------------------------
__global__ __launch_bounds__(256) void head_kernel(
    const float* __restrict__ x, const float* __restrict__ lnw,
    const float* __restrict__ lnb, const float* __restrict__ W,
    const float* __restrict__ hb, float* __restrict__ out) {
  const int wave = threadIdx.x >> 5, lane = threadIdx.x & 31;
  const int row = blockIdx.x * 8 + wave;
  if (row >= NTOK) return;
  const float* xr = x + (size_t)row * 128;
  float v[4], s = 0.f, s2 = 0.f;
#pragma unroll
  for (int c = 0; c < 4; ++c) {
    v[c] = xr[lane + 32 * c];
    s += v[c]; s2 += v[c] * v[c];
  }
  s = wred_sum(s); s2 = wred_sum(s2);
  const float mu = s * (1.0f / 128.0f);
  const float var = s2 * (1.0f / 128.0f) - mu * mu;
  const float rs = rsqrtf(var + 1e-5f);
  float d0 = 0.f, d1 = 0.f, d2 = 0.f;
#pragma unroll
  for (int c = 0; c < 4; ++c) {
    const int j = lane + 32 * c;
    const float nv = (v[c] - mu) * rs * lnw[j] + lnb[j];
    d0 += nv * W[j * 3 + 0];
    d1 += nv * W[j * 3 + 1];
    d2 += nv * W[j * 3 + 2];
  }
  d0 = wred_sum(d0); d1 = wred_sum(d1); d2 = wred_sum(d2);
  if (lane == 0) {
    out[row * 3 + 0] = d0 + hb[0];
    out[row * 3 + 1] = d1 + hb[1];
    out[row * 3 + 2] = d2 + hb[2];
  }
}

// ---------------------------------------------------------------------------
// Host-side orchestration
// ---------------------------------------------------------------------------
static inline void launch_gemm(const float* A, int lda, int arowmod,
                               const float* kscale,
                               const float* B, int ldb,
                               const float* bias,
                               const float* extra, const float* extra2,
                               float* C, int M, int K, int Ncols, int epi,
                               hipStream_t s) {
  const int grid = (M / 128) * (Ncols / 16);
  gemm_epi<<<grid, 256, 0, s>>>(A, lda, arowmod, kscale, B, ldb, bias,
                                extra, extra2, C, M, K, Ncols, epi);
}

extern "C" void kernel_launch(void* const* d_in, const int* in_sizes, int n_in,
                              void* d_out, int out_size, void* d_ws, size_t ws_size,
                              hipStream_t stream) {
  (void)in_sizes; (void)n_in; (void)out_size; (void)ws_size;

  const float* a_in  = (const float*)d_in[0];   // [512,768]
  const float* q_in  = (const float*)d_in[1];   // [1536,128]
  const float* c_in  = (const float*)d_in[2];   // [1536,128]
  const float* p_in  = (const float*)d_in[3];   // [1536,1536,16]
  const float* w_a          = (const float*)d_in[4];
  const float* attn_s_ln_w  = (const float*)d_in[5];
  const float* attn_scale_w = (const float*)d_in[6];
  const float* attn_scale_b = (const float*)d_in[7];
  const float* attn_shift_w = (const float*)d_in[8];
  const float* wq           = (const float*)d_in[9];
  const float* bq           = (const float*)d_in[10];
  const float* wk           = (const float*)d_in[11];
  const float* wv           = (const float*)d_in[12];
  const float* pair_ln_w    = (const float*)d_in[13];
  const float* pair_ln_b    = (const float*)d_in[14];
  const float* wb           = (const float*)d_in[15];
  const float* wg           = (const float*)d_in[16];
  const float* wo           = (const float*)d_in[17];
  const float* attn_gate_w  = (const float*)d_in[18];
  const float* attn_gate_b  = (const float*)d_in[19];
  const float* tr_s_ln_w    = (const float*)d_in[20];
  const float* tr_scale_w   = (const float*)d_in[21];
  const float* tr_scale_b   = (const float*)d_in[22];
  const float* tr_shift_w   = (const float*)d_in[23];
  const float* w1           = (const float*)d_in[24];
  const float* w2           = (const float*)d_in[25];
  const float* w3           = (const float*)d_in[26];
  const float* tr_gate_w    = (const float*)d_in[27];
  const float* tr_gate_b    = (const float*)d_in[28];
  const float* head_ln_w    = (const float*)d_in[29];
  const float* head_ln_b    = (const float*)d_in[30];
  const float* head_w       = (const float*)d_in[31];
  const float* head_b       = (const float*)d_in[32];

  // Workspace layout (floats)
  const size_t NQE = (size_t)NTOK * 128;
  float* ws    = (float*)d_ws;
  float* qbuf  = ws;             // current residual stream
  float* cn    = ws + 1 * NQE;   // LN0(c), shared by all blocks
  float* qn    = ws + 2 * NQE;   // LN0(qbuf)
  float* xn    = ws + 3 * NQE;   // adaLN output
  float* qh_s  = ws + 4 * NQE;
  float* kh_s  = ws + 5 * NQE;
  float* vh_s  = ws + 6 * NQE;
  float* gh_s  = ws + 7 * NQE;
  float* o_s   = ws + 8 * NQE;
  float* gate  = ws + 9 * NQE;
  float* battn = ws + 10 * NQE;
  float* tmp1  = ws + 11 * NQE;  // [N,256]
  float* tmp2  = ws + 13 * NQE;  // [N,256]
  float* biasb = ws + 15 * NQE;  // [N,128,4]

  const int EWG = (NTOK * 128) / 256; // 768 blocks for per-pair kernels
  const float* Z = nullptr;

  // q0 = tile(a,(3,1)) @ w_a + q
  launch_gemm(a_in, A_EMB, T_TOK, Z, w_a, Q_EMB, Z, q_in, Z, qbuf,
              NTOK, A_EMB, Q_EMB, EPI_ADD, stream);
  // cn = LN0(c) (shared by every block's adaLN)
  ln0_rows<<<NTOK / 8, 256, 0, stream>>>(c_in, cn);

  for (int i = 0; i < NBLK; ++i) {
    const int mo  = i * 128 * 128;
    const int mo2 = i * 128 * 256;
    const int vo  = i * 128;

    ln0_rows<<<NTOK / 8, 256, 0, stream>>>(qbuf, qn);

    // --- AttentionPairBias: adaLN (diag(s_ln_w) folded into B staging) ---
    launch_gemm(cn, 128, 0, attn_s_ln_w + vo, attn_scale_w + mo, 128,
                attn_scale_b + vo, Z, Z, tmp1, NTOK, 128, 128, EPI_NONE, stream);
    launch_gemm(cn, 128, 0, attn_s_ln_w + vo, attn_shift_w + mo, 128,
                Z, tmp1, qn, xn, NTOK, 128, 128, EPI_ADA, stream);

    // q/k/v/g projections
    launch_gemm(xn, 128, 0, Z, wq + mo, 128, bq + vo, Z, Z, qh_s, NTOK, 128, 128, EPI_NONE, stream);
    launch_gemm(xn, 128, 0, Z, wk + mo, 128, Z, Z, Z, kh_s, NTOK, 128, 128, EPI_NONE, stream);
    launch_gemm(xn, 128, 0, Z, wv + mo, 128, Z, Z, Z, vh_s, NTOK, 128, 128, EPI_NONE, stream);
    launch_gemm(xn, 128, 0, Z, wg + mo, 128, Z, Z, Z, gh_s, NTOK, 128, 128, EPI_NONE, stream);

    // windowed pair bias + fused attention
    pair_bias_kernel<<<EWG, 256, 0, stream>>>(p_in, pair_ln_w + i * 16,
                                              pair_ln_b + i * 16, wb + i * 64, biasb);
    attn_kernel<<<dim3(NTOK / NQ, NHEAD), 256, 0, stream>>>(qh_s, kh_s, vh_s,
                                                            gh_s, biasb, o_s);

    // output projection + conditioned gate
    launch_gemm(c_in, 128, 0, Z, attn_gate_w + mo, 128, attn_gate_b + vo,
                Z, Z, gate, NTOK, 128, 128, EPI_SIGMOID, stream);
    launch_gemm(o_s, 128, 0, Z, wo + mo, 128, Z, gate, Z,
                battn, NTOK, 128, 128, EPI_MUL, stream);

    // --- ConditionedTransitionBlock ---
    launch_gemm(cn, 128, 0, tr_s_ln_w + vo, tr_scale_w + mo, 128,
                tr_scale_b + vo, Z, Z, tmp1, NTOK, 128, 128, EPI_NONE, stream);
    launch_gemm(cn, 128, 0, tr_s_ln_w + vo, tr_shift_w + mo, 128,
                Z, tmp1, qn, xn, NTOK, 128, 128, EPI_ADA, stream);

    launch_gemm(xn, 128, 0, Z, w1 + mo2, 256, Z, Z, Z,
                tmp1, NTOK, 128, 256, EPI_SILU, stream);
    launch_gemm(xn, 128, 0, Z, w2 + mo2, 256, Z, tmp1, Z,
                tmp2, NTOK, 128, 256, EPI_MUL, stream);
    launch_gemm(c_in, 128, 0, Z, tr_gate_w + mo, 128, tr_gate_b + vo,
                Z, Z, gate, NTOK, 128, 128, EPI_SIGMOID, stream);
    // residual update fused: q = gate * (h @ w3) + battn
    launch_gemm(tmp2, 256, 0, Z, w3 + mo2, 128, Z, gate, battn,
                qbuf, NTOK, 256, 128, EPI_MULADD, stream);
  }

  head_kernel<<<NTOK / 8, 256, 0, stream>>>(qbuf, head_ln_w, head_ln_b,
                                            head_w, head_b, (float*)d_out);
}